// SimpleRoIAlign_23845658427770
// MI455X (gfx1250) — compile-verified
//
#include <hip/hip_runtime.h>
#include <hip/hip_bf16.h>

typedef __attribute__((ext_vector_type(2))) float v2f;
typedef __attribute__((ext_vector_type(8))) float v8f;

#define C_DIM 256
#define H_DIM 128
#define W_DIM 128
#define H_OUT 14
#define W_OUT 14
#define SCALE 0.25f
#define SEG   48          // column window per feature row
#define SEGP  49          // LDS pad (odd stride -> conflict-free column reads)
#define CHUNK 128         // channels per LDS staging pass
#define KTOT  96          // 2 rows * SEG

__global__ __launch_bounds__(256) void roi_align_wmma_kernel(
    const float* __restrict__ feats,   // (8,256,128,128)
    const float* __restrict__ rois,    // (1024,5): img, x1, y1, x2, y2
    float* __restrict__ out)           // (1024,256,14,14)
{
  __shared__ float ldsF[2][CHUNK][SEGP];   // two row segments, per-channel
  __shared__ float ldsB[KTOT][16];         // interpolation matrix (96 x 14, padded to 16)

  const int roi   = blockIdx.x;
  const int tid   = threadIdx.x;
  const int lane  = tid & 31;
  const int wave  = tid >> 5;              // 0..7 (wave32)
  const int half  = (lane >> 4) & 1;       // lane group within wave
  const int idx16 = lane & 15;

  // wave-uniform roi data (broadcast loads, L2-hot)
  const int   img = (int)rois[roi * 5 + 0];
  const float bx1 = rois[roi * 5 + 1];
  const float by1 = rois[roi * 5 + 2];
  const float bx2 = rois[roi * 5 + 3];
  const float by2 = rois[roi * 5 + 4];

  // x sampling window base (row-independent): xs = floor(px of sample 0)
  const float xabs0 = (0.5f / W_OUT) * (bx2 - bx1) + bx1;
  const float fpx0  = xabs0 * SCALE - 0.5f;
  const int   xs    = (int)floorf(fpx0);

  for (int orow = 0; orow < H_OUT; ++orow) {
    // y for this output row (uniform across the row)
    const float yabs = ((orow + 0.5f) / H_OUT) * (by2 - by1) + by1;
    const float fpy  = yabs * SCALE - 0.5f;
    const float y0f  = floorf(fpy);
    const float fy   = fpy - y0f;
    const int   y0   = (int)y0f;
    const int   y1   = y0 + 1;

    __syncthreads();                               // prior row's B reads complete
    for (int t = tid; t < KTOT * 16; t += 256) (&ldsB[0][0])[t] = 0.f;
    __syncthreads();

    if (tid < W_OUT) {                             // build sparse interp matrix B
      const float xabs = ((tid + 0.5f) / W_OUT) * (bx2 - bx1) + bx1;
      const float fpx  = xabs * SCALE - 0.5f;
      const float x0f  = floorf(fpx);
      const float fx   = fpx - x0f;
      int j0 = (int)x0f - xs;
      j0 = j0 < 0 ? 0 : (j0 > SEG - 2 ? SEG - 2 : j0);
      ldsB[j0        ][tid] = (1.f - fx) * (1.f - fy);  // (y0, x0)
      ldsB[j0 + 1    ][tid] =        fx  * (1.f - fy);  // (y0, x1)
      ldsB[SEG + j0  ][tid] = (1.f - fx) * fy;          // (y1, x0)
      ldsB[SEG + j0+1][tid] =        fx  * fy;          // (y1, x1)
    }

    for (int chunk = 0; chunk < C_DIM / CHUNK; ++chunk) {
      const int cb = chunk * CHUNK;
      __syncthreads();                             // B ready / prior ldsF reads done

      // Stage two contiguous feature-row segments for 128 channels (coalesced).
      // Zero-fill outside the tensor == reference's zero-padded clamp.
      for (int t = tid; t < 2 * CHUNK * SEG; t += 256) {
        const int j    = t % SEG;
        const int rc   = t / SEG;
        const int c    = rc & (CHUNK - 1);
        const int rsel = rc >> 7;
        const int y    = rsel ? y1 : y0;
        const int col  = xs + j;
        float v = 0.f;
        if ((unsigned)col < (unsigned)W_DIM && (unsigned)y < (unsigned)H_DIM) {
          v = feats[(((long)(img * C_DIM + cb + c)) * H_DIM + y) * W_DIM + col];
        }
        ldsF[rsel][c][j] = v;
      }
      __syncthreads();

      // Each wave: 16-channel x 16-pixel tile, K=96 in 24 f32 WMMA k-steps.
      const int clBase = wave * 16;
      v8f acc = {};
#pragma unroll
      for (int ks = 0; ks < KTOT / 4; ++ks) {
        const int k0 = ks * 4;
        const int kA = k0 + 2 * half;              // this lane group's K pair
        const int rs = (kA >= SEG) ? 1 : 0;
        const int jA = kA - (rs ? SEG : 0);
        v2f a, b;
        a.x = ldsF[rs][clBase + idx16][jA];        // A: M=idx16, K=kA
        a.y = ldsF[rs][clBase + idx16][jA + 1];    //    K=kA+1
        b.x = ldsB[kA][idx16];                     // B: N=idx16, K=kA
        b.y = ldsB[kA + 1][idx16];
        acc = __builtin_amdgcn_wmma_f32_16x16x4_f32(
            /*neg_a=*/false, a, /*neg_b=*/false, b,
            /*c_mod=*/(short)0, acc, /*reuse_a=*/false, /*reuse_b=*/false);
      }

      // D layout: lane l, vgpr r -> M = r + 8*(l/16), N = l%16
      if (idx16 < W_OUT) {
#pragma unroll
        for (int r2 = 0; r2 < 8; ++r2) {
          const int ch = cb + clBase + r2 + 8 * half;
          out[(((long)roi * C_DIM + ch) * H_OUT + orow) * W_OUT + idx16] = acc[r2];
        }
      }
    }
  }
}

extern "C" void kernel_launch(void* const* d_in, const int* in_sizes, int n_in,
                              void* d_out, int out_size, void* d_ws, size_t ws_size,
                              hipStream_t stream) {
  (void)in_sizes; (void)n_in; (void)out_size; (void)d_ws; (void)ws_size;
  const float* feats = (const float*)d_in[0];
  const float* rois  = (const float*)d_in[1];
  float* out = (float*)d_out;
  roi_align_wmma_kernel<<<1024, 256, 0, stream>>>(feats, rois, out);
}